// TrajectoryLSTMSegmentorV2_77687368450094
// MI455X (gfx1250) — compile-verified
//
#include <hip/hip_runtime.h>
#include <hip/hip_bf16.h>

typedef __attribute__((ext_vector_type(2))) float v2f;
typedef __attribute__((ext_vector_type(8))) float v8f;

#define C      256
#define BSEG   8
#define NCLS   20
#define NPAD   32
#define HID    512

// workspace layout (floats)
#define WS_SUMS  0                 // [8][256] segment sums
#define WS_WPAD  2048              // [32][256] zero-padded W_head
#define WS_DELTA (2048 + 32*256)   // [8][32] per-segment logit bias (incl b_head)

#define ROWS_PER_CHUNK 64
#define THREADS_A      128
#define CPB            8           // chunks per block in main kernel

#if __has_builtin(__builtin_amdgcn_wmma_f32_16x16x4_f32)
#define HAVE_WMMA_F32 1
#else
#define HAVE_WMMA_F32 0
#endif

// ---------------------------------------------------------------------------
// Kernel 0: zero segment sums, build padded B matrix Wpad[32][256]
// ---------------------------------------------------------------------------
__global__ void __launch_bounds__(256) prep_kernel(const float* __restrict__ W_head,
                                                   float* __restrict__ ws) {
    int i = blockIdx.x * blockDim.x + threadIdx.x;
    if (i < BSEG * C) ws[WS_SUMS + i] = 0.0f;
    if (i < NPAD * C) {
        int n = i >> 8, k = i & (C - 1);
        ws[WS_WPAD + i] = (n < NCLS) ? W_head[n * C + k] : 0.0f;
    }
}

__device__ __forceinline__ int seg_of(int row, const int* off) {
    int s = 0;
#pragma unroll
    for (int b = 0; b < BSEG - 1; ++b) s += (row >= off[b]) ? 1 : 0;
    return s;
}

// ---------------------------------------------------------------------------
// Kernel 1: fused  out = feat @ Wpad^T  (WMMA f32)  +  per-segment col sums
// block = 128 threads (4 waves); each block processes CPB chunks of 64 rows
// ---------------------------------------------------------------------------
__global__ void __launch_bounds__(THREADS_A) head_gemm_pool_kernel(
    const float* __restrict__ feat, const int* __restrict__ offset,
    const float* __restrict__ wpad, float* __restrict__ sums,
    float* __restrict__ out, int nChunks) {
    __shared__ float tile[ROWS_PER_CHUNK * C];  // 64 KB, XOR-swizzled rows

    const int t  = threadIdx.x;
    const int c0 = blockIdx.x * CPB;
    if (c0 >= nChunks) return;
    const int c1 = min(c0 + CPB, nChunks);

    int off[BSEG];
#pragma unroll
    for (int b = 0; b < BSEG; ++b) off[b] = offset[b];

    const int sA      = seg_of(c0 * ROWS_PER_CHUNK, off);
    const int offEndA = off[sA];

    // staging thread t always loads columns [colBase, colBase+4)
    const int colBase  = (t * 4) & (C - 1);
    const int rowPar   = t >> 6;  // 0 or 1 (row parity)
    float4 accA = make_float4(0.f, 0.f, 0.f, 0.f);
    float4 accB = make_float4(0.f, 0.f, 0.f, 0.f);

    // wmma lane mapping
    const int wv = t >> 5, ln = t & 31, m = ln & 15, hi = ln >> 4;
    const int xorm = m << 2;
    const float* wp0 = wpad + (size_t)m * C;         // B rows n = 0..15
    const float* wp1 = wpad + (size_t)(16 + m) * C;  // B rows n = 16..31

    for (int ch = c0; ch < c1; ++ch) {
        const int row0 = ch * ROWS_PER_CHUNK;

        // ---- stage 64x256 tile into LDS (coalesced) + pooling accumulate ----
#pragma unroll 8
        for (int j = 0; j < 32; ++j) {
            const int lr = 2 * j + rowPar;         // local row 0..63
            const int r  = row0 + lr;
            const float4 g = *(const float4*)(feat + (size_t)r * C + colBase);
            const bool inA = (r < offEndA);
            accA.x += inA ? g.x : 0.f;  accB.x += inA ? 0.f : g.x;
            accA.y += inA ? g.y : 0.f;  accB.y += inA ? 0.f : g.y;
            accA.z += inA ? g.z : 0.f;  accB.z += inA ? 0.f : g.z;
            accA.w += inA ? g.w : 0.f;  accB.w += inA ? 0.f : g.w;
            const int ksw = colBase ^ ((lr & 15) << 2);  // bank-conflict swizzle
            *(float4*)(tile + lr * C + ksw) = g;
        }
        __syncthreads();

        // ---- WMMA: 16x16x4_f32, K=256, two 16-wide column tiles ----
        const float* arow = tile + (size_t)(wv * 16 + m) * C;
        v8f d0 = {};
        v8f d1 = {};
#pragma unroll 8
        for (int kb = 0; kb < C; kb += 4) {
            const v2f a  = *(const v2f*)(arow + ((kb ^ xorm) + 2 * hi));
            const v2f b0 = *(const v2f*)(wp0 + kb + 2 * hi);
            const v2f b1 = *(const v2f*)(wp1 + kb + 2 * hi);
#if HAVE_WMMA_F32
            d0 = __builtin_amdgcn_wmma_f32_16x16x4_f32(false, a, false, b0,
                                                       (short)0, d0, false, false);
            d1 = __builtin_amdgcn_wmma_f32_16x16x4_f32(false, a, false, b1,
                                                       (short)0, d1, false, false);
#else
            // compile-insurance fallback (scalar FMA; layout-inexact)
            d0[0] += a.x * b0.x + a.y * b0.y;
            d1[0] += a.x * b1.x + a.y * b1.y;
#endif
        }

        // ---- store D: lane (hi,m), VGPR r -> out[M = r + 8*hi][N = m (+16)] ----
        const int growbase = row0 + wv * 16 + hi * 8;
#pragma unroll
        for (int r = 0; r < 8; ++r) {
            const int gr = growbase + r;
            out[(size_t)gr * NCLS + m] = d0[r];
            if (m < 4) out[(size_t)gr * NCLS + 16 + m] = d1[r];
        }
        __syncthreads();
    }

    // ---- pooling atomics (<= 2 segments per block) ----
    atomicAdd(&sums[sA * C + colBase + 0], accA.x);
    atomicAdd(&sums[sA * C + colBase + 1], accA.y);
    atomicAdd(&sums[sA * C + colBase + 2], accA.z);
    atomicAdd(&sums[sA * C + colBase + 3], accA.w);
    if (sA + 1 < BSEG && c1 * ROWS_PER_CHUNK > offEndA) {
        atomicAdd(&sums[(sA + 1) * C + colBase + 0], accB.x);
        atomicAdd(&sums[(sA + 1) * C + colBase + 1], accB.y);
        atomicAdd(&sums[(sA + 1) * C + colBase + 2], accB.z);
        atomicAdd(&sums[(sA + 1) * C + colBase + 3], accB.w);
    }
}

// ---------------------------------------------------------------------------
// Kernel 2: pooled -> LSTM cell (h0=c0=0) -> ctx -> delta = ctx@W_head^T + b
// one block per segment
// ---------------------------------------------------------------------------
__global__ void __launch_bounds__(256) lstm_ctx_kernel(
    const float* __restrict__ sums, const int* __restrict__ offset,
    const float* __restrict__ W_ih, const float* __restrict__ b_ih,
    const float* __restrict__ b_hh, const float* __restrict__ W_proj,
    const float* __restrict__ b_proj, const float* __restrict__ W_head,
    const float* __restrict__ b_head, float* __restrict__ delta) {
    __shared__ float pooled[C];
    __shared__ float hbuf[HID];
    __shared__ float ctx[C];
    const int b = blockIdx.x, t = threadIdx.x;

    const int start = (b == 0) ? 0 : offset[b - 1];
    const float inv = 1.0f / (float)(offset[b] - start);
    pooled[t] = sums[b * C + t] * inv;
    __syncthreads();

    // gate order i, f, g, o ; f unused with zero cell state
    for (int hh = t; hh < HID; hh += 256) {
        float gi = b_ih[hh] + b_hh[hh];
        float gg = b_ih[2 * HID + hh] + b_hh[2 * HID + hh];
        float go = b_ih[3 * HID + hh] + b_hh[3 * HID + hh];
        const float* wi = W_ih + (size_t)hh * C;
        const float* wg = W_ih + (size_t)(2 * HID + hh) * C;
        const float* wo = W_ih + (size_t)(3 * HID + hh) * C;
        for (int c = 0; c < C; ++c) {
            const float p = pooled[c];
            gi += p * wi[c];
            gg += p * wg[c];
            go += p * wo[c];
        }
        const float ig = 1.f / (1.f + __expf(-gi));
        const float og = 1.f / (1.f + __expf(-go));
        const float cs = ig * tanhf(gg);
        hbuf[hh] = og * tanhf(cs);
    }
    __syncthreads();

    {
        float acc = b_proj[t];
        const float* wr = W_proj + (size_t)t * HID;
        for (int k = 0; k < HID; ++k) acc += wr[k] * hbuf[k];
        ctx[t] = acc;
    }
    __syncthreads();

    if (t < NPAD) {
        float v = 0.f;
        if (t < NCLS) {
            v = b_head[t];
            const float* wr = W_head + (size_t)t * C;
            for (int k = 0; k < C; ++k) v += wr[k] * ctx[k];
        }
        delta[b * NPAD + t] = v;
    }
}

// ---------------------------------------------------------------------------
// Kernel 3: out[i, n] += delta[seg(i), n]
// ---------------------------------------------------------------------------
__global__ void __launch_bounds__(256) apply_ctx_kernel(
    float* __restrict__ out, const float* __restrict__ delta,
    const int* __restrict__ offset, long long total) {
    const long long e = (long long)blockIdx.x * blockDim.x + threadIdx.x;
    if (e >= total) return;
    const int i = (int)(e / NCLS);
    const int n = (int)(e - (long long)i * NCLS);
    int s = 0;
#pragma unroll
    for (int b = 0; b < BSEG - 1; ++b) s += (i >= offset[b]) ? 1 : 0;
    out[e] += delta[s * NPAD + n];
}

// ---------------------------------------------------------------------------
extern "C" void kernel_launch(void* const* d_in, const int* in_sizes, int n_in,
                              void* d_out, int out_size, void* d_ws, size_t ws_size,
                              hipStream_t stream) {
    const float* feat   = (const float*)d_in[0];
    const int*   offset = (const int*)d_in[1];
    const float* W_ih   = (const float*)d_in[2];
    const float* b_ih   = (const float*)d_in[3];
    // d_in[4] = W_hh (unused: h0 = 0)
    const float* b_hh   = (const float*)d_in[5];
    const float* W_proj = (const float*)d_in[6];
    const float* b_proj = (const float*)d_in[7];
    const float* W_head = (const float*)d_in[8];
    const float* b_head = (const float*)d_in[9];
    float* out = (float*)d_out;
    float* ws  = (float*)d_ws;

    const int N = in_sizes[0] / C;                  // 1,000,000
    const int nChunks = N / ROWS_PER_CHUNK;         // 15625
    const int nbMain  = (nChunks + CPB - 1) / CPB;  // 1954

    float* sums  = ws + WS_SUMS;
    float* wpad  = ws + WS_WPAD;
    float* delta = ws + WS_DELTA;

    prep_kernel<<<(NPAD * C + 255) / 256, 256, 0, stream>>>(W_head, ws);

    head_gemm_pool_kernel<<<nbMain, THREADS_A, 0, stream>>>(
        feat, offset, wpad, sums, out, nChunks);

    lstm_ctx_kernel<<<BSEG, 256, 0, stream>>>(
        sums, offset, W_ih, b_ih, b_hh, W_proj, b_proj, W_head, b_head, delta);

    const long long total = (long long)N * NCLS;
    const int nbApply = (int)((total + 255) / 256);
    apply_ctx_kernel<<<nbApply, 256, 0, stream>>>(out, delta, offset, total);
}